// T2toT2_82738249990284
// MI455X (gfx1250) — compile-verified
//
#include <hip/hip_runtime.h>
#include <hip/hip_bf16.h>
#include <math.h>

typedef __attribute__((ext_vector_type(2))) float v2f;
typedef __attribute__((ext_vector_type(8))) float v8f;

#define Ldim 48
#define Bdim 16
#define Cdim 128
#define Edim 128
#define MROWS (Ldim * Ldim * Bdim)      // 36864
#define ROWTILES (MROWS / 16)           // 2304
#define LDA 132                         // padded A-tile row stride (words)
#define LDB 136                         // padded W-tile row stride (words)

// ---------------------------------------------------------------------------
// GEMM1: H = silu( BN(T) @ w1 + b1 ), fp32 WMMA 16x16x4, persistent blocks.
// block = 256 threads = 8 waves; wave w owns cols [16w, 16w+16).
// ---------------------------------------------------------------------------
__global__ void gemm1_bn_silu_kernel(const float* __restrict__ Tg,
                                     const float* __restrict__ rm,
                                     const float* __restrict__ rv,
                                     const float* __restrict__ bw,
                                     const float* __restrict__ bb,
                                     const float* __restrict__ w1,
                                     const float* __restrict__ b1,
                                     float* __restrict__ H) {
    extern __shared__ float smem[];
    float* Wl = smem;                    // 128 * LDB
    float* Al = Wl + Cdim * LDB;         // 16 * LDA
    float* sc = Al + 16 * LDA;           // 128
    float* sh = sc + Cdim;               // 128

    const int tid  = threadIdx.x;
    const int lane = tid & 31;
    const int wv   = tid >> 5;
    const int n0   = wv * 16;
    const int hi   = lane >> 4;
    const int nl   = lane & 15;

    if (tid < Cdim) {
        float s = bw[tid] * rsqrtf(rv[tid] + 1e-5f);
        sc[tid] = s;
        sh[tid] = bb[tid] - rm[tid] * s;
    }
    // stage w1 [128 x 128] into LDS (float4 chunks)
    for (int c = tid; c < Cdim * 32; c += 256) {
        int k = c >> 5, n4 = (c & 31) << 2;
        *(float4*)(Wl + k * LDB + n4) = *(const float4*)(w1 + k * Edim + n4);
    }
    __syncthreads();

    for (int rt = blockIdx.x; rt < ROWTILES; rt += gridDim.x) {
        // stage 16x128 A tile with fused BN affine
        #pragma unroll
        for (int q = 0; q < 8; ++q) {
            int idx = tid + 256 * q;               // 0..2047
            int r = idx >> 7, k = idx & 127;
            float x = Tg[(size_t)(rt * 16 + r) * Cdim + k];
            Al[r * LDA + k] = x * sc[k] + sh[k];
        }
        __syncthreads();

        v8f acc = {0.f, 0.f, 0.f, 0.f, 0.f, 0.f, 0.f, 0.f};
        #pragma unroll
        for (int kk = 0; kk < 32; ++kk) {
            int k0 = kk * 4 + hi * 2;
            v2f a = *(const v2f*)(Al + nl * LDA + k0);
            v2f bfr;
            bfr.x = Wl[k0 * LDB + n0 + nl];
            bfr.y = Wl[(k0 + 1) * LDB + n0 + nl];
            acc = __builtin_amdgcn_wmma_f32_16x16x4_f32(
                false, a, false, bfr, (short)0, acc, false, false);
        }

        float bias = b1[n0 + nl];
        #pragma unroll
        for (int v = 0; v < 8; ++v) {
            int r = v + 8 * hi;
            float x = acc[v] + bias;
            float hv = x / (1.0f + __expf(-x));     // silu
            H[(size_t)(rt * 16 + r) * Edim + n0 + nl] = hv;
        }
        __syncthreads();
    }
}

// ---------------------------------------------------------------------------
// Aggregations: diag[x,b,e], row_agg[x,b,e] (sum over i), col_agg[x,b,e] (sum over j)
// ---------------------------------------------------------------------------
__global__ void agg1_kernel(const float* __restrict__ H, const int* __restrict__ mask,
                            float* __restrict__ diag, float* __restrict__ rowg,
                            float* __restrict__ colg) {
    int idx = blockIdx.x * blockDim.x + threadIdx.x;
    if (idx >= Ldim * Bdim * Edim) return;
    int e = idx & 127, rem = idx >> 7;
    int b = rem & 15, x = rem >> 4;
    float rs = 0.f, cs = 0.f;
    for (int t = 0; t < Ldim; ++t) {
        int mcol = (x * Ldim + t) * Bdim + b;       // H[x, t, b] -> col_agg (sum over j)
        if (mask[mcol]) cs += H[(size_t)mcol * Edim + e];
        int mrow = (t * Ldim + x) * Bdim + b;       // H[t, x, b] -> row_agg (sum over i)
        if (mask[mrow]) rs += H[(size_t)mrow * Edim + e];
    }
    diag[idx] = H[(size_t)((x * Ldim + x) * Bdim + b) * Edim + e];
    rowg[idx] = rs * (1.0f / 60.0f);
    colg[idx] = cs * (1.0f / 60.0f);
}

// ---------------------------------------------------------------------------
// diag_agg[b,e], all_agg[b,e]
// ---------------------------------------------------------------------------
__global__ void agg2_kernel(const float* __restrict__ H, const int* __restrict__ mask,
                            const float* __restrict__ rowg,
                            float* __restrict__ dagg, float* __restrict__ aagg) {
    int idx = blockIdx.x * blockDim.x + threadIdx.x;
    if (idx >= Bdim * Edim) return;
    int e = idx & 127, b = idx >> 7;
    float ds = 0.f, as = 0.f;
    for (int t = 0; t < Ldim; ++t) {
        int md = (t * Ldim + t) * Bdim + b;
        if (mask[md]) ds += H[(size_t)md * Edim + e];
        as += rowg[(size_t)(t * Bdim + b) * Edim + e];
    }
    dagg[idx] = ds * (1.0f / 60.0f);
    aagg[idx] = as * (1.0f / 60.0f);   // rowg already /60 -> total /3600
}

// ---------------------------------------------------------------------------
// Small matmuls against w2 blocks: P, Q, D per (i,b); R per (b).
// W_p[k][n] = w2[(p*128 + k)*128 + n]
// ---------------------------------------------------------------------------
__global__ void small_mm_kernel(const float* __restrict__ diag, const float* __restrict__ rowg,
                                const float* __restrict__ colg, const float* __restrict__ dagg,
                                const float* __restrict__ aagg, const float* __restrict__ w2,
                                const float* __restrict__ b2,
                                float* __restrict__ P, float* __restrict__ Q,
                                float* __restrict__ Dd, float* __restrict__ R) {
    int idx = blockIdx.x * blockDim.x + threadIdx.x;
    if (idx >= Ldim * Bdim * Cdim) return;
    int n = idx & 127, rem = idx >> 7;
    int b = rem & 15, i = rem >> 4;
    const size_t base = (size_t)(i * Bdim + b) * Edim;
    float p = 0.f, q = 0.f, dd = 0.f;
    for (int k = 0; k < Edim; ++k) {
        float d = diag[base + k];
        float r = rowg[base + k];
        float c = colg[base + k];
        float g = dagg[b * Edim + k];
        p  += d * w2[(size_t)(2  * 128 + k) * Cdim + n]
            + r * w2[(size_t)(13 * 128 + k) * Cdim + n]
            + c * w2[(size_t)(12 * 128 + k) * Cdim + n];
        q  += d * w2[(size_t)(1  * 128 + k) * Cdim + n]
            + r * w2[(size_t)(9  * 128 + k) * Cdim + n]
            + c * w2[(size_t)(10 * 128 + k) * Cdim + n];
        dd += d * w2[(size_t)(0  * 128 + k) * Cdim + n]
            + r * w2[(size_t)(5  * 128 + k) * Cdim + n]
            + c * w2[(size_t)(6  * 128 + k) * Cdim + n]
            + g * w2[(size_t)(7  * 128 + k) * Cdim + n];
    }
    P[idx] = p; Q[idx] = q; Dd[idx] = dd;
    if (i == 0) {
        float rr = b2[n];
        for (int k = 0; k < Edim; ++k) {
            rr += dagg[b * Edim + k] * w2[(size_t)(8 * 128 + k) * Cdim + n]
                + aagg[b * Edim + k] * (w2[(size_t)(11 * 128 + k) * Cdim + n]
                                      + w2[(size_t)(14 * 128 + k) * Cdim + n]);
        }
        R[b * Cdim + n] = rr;
    }
}

// ---------------------------------------------------------------------------
// GEMM2 + epilogue:
// out[i,j,b] = H[i,j,b]@W4 + H[j,i,b]@W3 + P[i,b] + Q[j,b] + R[b] + (i==j)*D[i,b]
// ---------------------------------------------------------------------------
__global__ void gemm2_final_kernel(const float* __restrict__ H, const float* __restrict__ w2,
                                   const float* __restrict__ P, const float* __restrict__ Q,
                                   const float* __restrict__ Dd, const float* __restrict__ R,
                                   float* __restrict__ out) {
    extern __shared__ float smem[];
    float* W4l = smem;                    // 128 * LDB
    float* W3l = W4l + Cdim * LDB;        // 128 * LDB
    float* A1  = W3l + Cdim * LDB;        // 16 * LDA
    float* A2  = A1 + 16 * LDA;           // 16 * LDA

    const int tid  = threadIdx.x;
    const int lane = tid & 31;
    const int wv   = tid >> 5;
    const int n0   = wv * 16;
    const int hi   = lane >> 4;
    const int nl   = lane & 15;

    for (int c = tid; c < Cdim * 32; c += 256) {
        int k = c >> 5, n4 = (c & 31) << 2;
        *(float4*)(W4l + k * LDB + n4) = *(const float4*)(w2 + (size_t)(4 * 128 + k) * Cdim + n4);
        *(float4*)(W3l + k * LDB + n4) = *(const float4*)(w2 + (size_t)(3 * 128 + k) * Cdim + n4);
    }
    __syncthreads();

    for (int rt = blockIdx.x; rt < ROWTILES; rt += gridDim.x) {
        int i = rt / Ldim, j = rt - i * Ldim;
        int base1 = rt * 16;                    // rows of H[i,j,:]
        int base2 = (j * Ldim + i) * 16;        // rows of H[j,i,:]
        #pragma unroll
        for (int q = 0; q < 8; ++q) {
            int idx = tid + 256 * q;
            int r = idx >> 7, k = idx & 127;
            A1[r * LDA + k] = H[(size_t)(base1 + r) * Edim + k];
            A2[r * LDA + k] = H[(size_t)(base2 + r) * Edim + k];
        }
        __syncthreads();

        v8f acc = {0.f, 0.f, 0.f, 0.f, 0.f, 0.f, 0.f, 0.f};
        #pragma unroll
        for (int kk = 0; kk < 32; ++kk) {
            int k0 = kk * 4 + hi * 2;
            v2f a = *(const v2f*)(A1 + nl * LDA + k0);
            v2f bfr;
            bfr.x = W4l[k0 * LDB + n0 + nl];
            bfr.y = W4l[(k0 + 1) * LDB + n0 + nl];
            acc = __builtin_amdgcn_wmma_f32_16x16x4_f32(
                false, a, false, bfr, (short)0, acc, false, false);
        }
        #pragma unroll
        for (int kk = 0; kk < 32; ++kk) {
            int k0 = kk * 4 + hi * 2;
            v2f a = *(const v2f*)(A2 + nl * LDA + k0);
            v2f bfr;
            bfr.x = W3l[k0 * LDB + n0 + nl];
            bfr.y = W3l[(k0 + 1) * LDB + n0 + nl];
            acc = __builtin_amdgcn_wmma_f32_16x16x4_f32(
                false, a, false, bfr, (short)0, acc, false, false);
        }

        int n = n0 + nl;
        bool onDiag = (i == j);
        #pragma unroll
        for (int v = 0; v < 8; ++v) {
            int bb = v + 8 * hi;                // batch index within tile
            float val = acc[v]
                      + P[(size_t)(i * Bdim + bb) * Cdim + n]
                      + Q[(size_t)(j * Bdim + bb) * Cdim + n]
                      + R[bb * Cdim + n];
            if (onDiag) val += Dd[(size_t)(i * Bdim + bb) * Cdim + n];
            out[(size_t)(base1 + bb) * Cdim + n] = val;
        }
        __syncthreads();
    }
}

// ---------------------------------------------------------------------------
extern "C" void kernel_launch(void* const* d_in, const int* in_sizes, int n_in,
                              void* d_out, int out_size, void* d_ws, size_t ws_size,
                              hipStream_t stream) {
    const float* T   = (const float*)d_in[0];
    const int*   msk = (const int*)  d_in[1];
    const float* rm  = (const float*)d_in[2];
    const float* rv  = (const float*)d_in[3];
    const float* bw  = (const float*)d_in[4];
    const float* bb  = (const float*)d_in[5];
    const float* w1  = (const float*)d_in[6];
    const float* b1  = (const float*)d_in[7];
    const float* w2  = (const float*)d_in[8];
    const float* b2  = (const float*)d_in[9];
    float* out = (float*)d_out;
    float* ws  = (float*)d_ws;

    const size_t nH   = (size_t)MROWS * Edim;           // 4,718,592
    const size_t nLBE = (size_t)Ldim * Bdim * Edim;     // 98,304
    const size_t nBE  = (size_t)Bdim * Edim;            // 2,048

    float* H    = ws;
    float* diag = H    + nH;
    float* rowg = diag + nLBE;
    float* colg = rowg + nLBE;
    float* dagg = colg + nLBE;
    float* aagg = dagg + nBE;
    float* P    = aagg + nBE;
    float* Q    = P    + nLBE;
    float* Dd   = Q    + nLBE;
    float* R    = Dd   + nLBE;

    const size_t shm1 = (size_t)(Cdim * LDB + 16 * LDA + 2 * Cdim) * sizeof(float);
    const size_t shm2 = (size_t)(2 * Cdim * LDB + 2 * 16 * LDA) * sizeof(float);

    gemm1_bn_silu_kernel<<<576, 256, shm1, stream>>>(T, rm, rv, bw, bb, w1, b1, H);
    agg1_kernel<<<(int)((nLBE + 255) / 256), 256, 0, stream>>>(H, msk, diag, rowg, colg);
    agg2_kernel<<<(int)((nBE + 255) / 256), 256, 0, stream>>>(H, msk, rowg, dagg, aagg);
    small_mm_kernel<<<(int)((nLBE + 255) / 256), 256, 0, stream>>>(
        diag, rowg, colg, dagg, aagg, w2, b2, P, Q, Dd, R);
    gemm2_final_kernel<<<576, 256, shm2, stream>>>(H, w2, P, Q, Dd, R, out);
}